// BahdanauAttention_2018634629612
// MI455X (gfx1250) — compile-verified
//
#include <hip/hip_runtime.h>
#include <math.h>

typedef __attribute__((ext_vector_type(2))) float v2f;
typedef __attribute__((ext_vector_type(8))) float v8f;

#define WMMA_F32_16x16x4(a, b, c) \
    __builtin_amdgcn_wmma_f32_16x16x4_f32(false, (a), false, (b), (short)0, (c), false, false)

// ---------------------------------------------------------------------------
// GEMM: Y (M x N) = X (M x K, row-major) @ W^T, where W is (N x K, row-major)
// One 16x16 output tile per wave; K-loop in steps of 4 via v_wmma_f32_16x16x4_f32.
// A frag: lane m=lane&15, kk=(lane>>4)*2 -> float2 from X row (contiguous).
// B frag: lane n=lane&15, same kk        -> float2 from W row (contiguous).
// ---------------------------------------------------------------------------
__global__ void gemm_xwT_wmma(const float* __restrict__ X, const float* __restrict__ W,
                              float* __restrict__ Y, int M, int N, int K) {
    const int wave = threadIdx.x >> 5;
    const int lane = threadIdx.x & 31;
    const int tile = blockIdx.x * (blockDim.x >> 5) + wave;
    const int ntn  = N >> 4;
    if (tile >= (M >> 4) * ntn) return;               // uniform per-wave
    const int tm   = (tile / ntn) << 4;
    const int tn   = (tile % ntn) << 4;
    const int half = lane >> 4;
    const int l    = lane & 15;
    const int kk   = half << 1;

    const float* arow = X + (size_t)(tm + l) * K;
    const float* brow = W + (size_t)(tn + l) * K;
    v8f acc = {};
    for (int k = 0; k < K; k += 4) {
        v2f a = *(const v2f*)(arow + k + kk);
        v2f b = *(const v2f*)(brow + k + kk);
        acc = WMMA_F32_16x16x4(a, b, acc);
    }
    const int mbase = tm + half * 8;
#pragma unroll
    for (int i = 0; i < 8; ++i)
        Y[(size_t)(mbase + i) * N + tn + l] = acc[i];
}

// ---------------------------------------------------------------------------
// Batched GEMM: C[b] (M x N) = A[b] (M x K, row-major) @ Bm[b] (K x N, row-major)
// Used for context = alpha @ encoder_outputs.  B frag: two strided scalar loads.
// ---------------------------------------------------------------------------
__global__ void gemm_ab_batched_wmma(const float* __restrict__ A, const float* __restrict__ Bm,
                                     float* __restrict__ Y, int M, int N, int K) {
    const int wave  = threadIdx.x >> 5;
    const int lane  = threadIdx.x & 31;
    const int batch = blockIdx.y;
    const int tile  = blockIdx.x * (blockDim.x >> 5) + wave;
    const int ntn   = N >> 4;
    if (tile >= (M >> 4) * ntn) return;
    const int tm   = (tile / ntn) << 4;
    const int tn   = (tile % ntn) << 4;
    const int half = lane >> 4;
    const int l    = lane & 15;
    const int kk   = half << 1;

    const float* Ab = A  + (size_t)batch * M * K;
    const float* Bb = Bm + (size_t)batch * K * N;
    float*       Yb = Y  + (size_t)batch * M * N;

    const float* arow = Ab + (size_t)(tm + l) * K;
    const int n = tn + l;
    v8f acc = {};
    for (int k = 0; k < K; k += 4) {
        v2f a = *(const v2f*)(arow + k + kk);
        v2f b;
        b.x = Bb[(size_t)(k + kk) * N + n];
        b.y = Bb[(size_t)(k + kk + 1) * N + n];
        acc = WMMA_F32_16x16x4(a, b, acc);
    }
    const int mbase = tm + half * 8;
#pragma unroll
    for (int i = 0; i < 8; ++i)
        Yb[(size_t)(mbase + i) * N + n] = acc[i];
}

// ---------------------------------------------------------------------------
// Output projection: out (512 x 256) = tanh( [query | context] (512 x 512) @ Wout^T )
// Concat handled as two K=256 accumulation passes (query then context) into one
// accumulator; Wout is (256 x 512) row-major.
// ---------------------------------------------------------------------------
__global__ void out_gemm_wmma(const float* __restrict__ Q, const float* __restrict__ Cc,
                              const float* __restrict__ Wout, float* __restrict__ Yout) {
    const int M = 512, N = 256, H = 256, F = 512;
    const int wave = threadIdx.x >> 5;
    const int lane = threadIdx.x & 31;
    const int tile = blockIdx.x * (blockDim.x >> 5) + wave;
    const int ntn  = N >> 4;
    if (tile >= (M >> 4) * ntn) return;
    const int tm   = (tile / ntn) << 4;
    const int tn   = (tile % ntn) << 4;
    const int half = lane >> 4;
    const int l    = lane & 15;
    const int kk   = half << 1;

    const float* qrow = Q    + (size_t)(tm + l) * H;
    const float* crow = Cc   + (size_t)(tm + l) * H;
    const float* wrow = Wout + (size_t)(tn + l) * F;

    v8f acc = {};
    for (int k = 0; k < H; k += 4) {                 // query half of concat
        v2f a = *(const v2f*)(qrow + k + kk);
        v2f b = *(const v2f*)(wrow + k + kk);
        acc = WMMA_F32_16x16x4(a, b, acc);
    }
    for (int k = 0; k < H; k += 4) {                 // context half of concat
        v2f a = *(const v2f*)(crow + k + kk);
        v2f b = *(const v2f*)(wrow + H + k + kk);
        acc = WMMA_F32_16x16x4(a, b, acc);
    }
    const int mbase = tm + half * 8;
#pragma unroll
    for (int i = 0; i < 8; ++i)
        Yout[(size_t)(mbase + i) * N + tn + l] = tanhf(acc[i]);
}

// ---------------------------------------------------------------------------
// Additive score + mask + softmax, one block per (b,t).
// e[s] = sum_h v[h]*tanh(pq[bt,h] + pe[b,s,h]); masked by src_len; softmax over S.
// 8 waves x 64 s-values; lanes stride over h (coalesced pe reads) + butterfly reduce.
// ---------------------------------------------------------------------------
__global__ void score_softmax_kernel(const float* __restrict__ pq,   // (B*T, H)
                                     const float* __restrict__ pe,   // (B*S, H)
                                     const float* __restrict__ v,    // (H)
                                     const int*   __restrict__ src_len,
                                     float* __restrict__ alpha) {    // (B*T, S)
    const int H = 256, S = 512, T = 128;
    const int bt   = blockIdx.x;         // 0..511
    const int b    = bt / T;
    const int tid  = threadIdx.x;        // 256 threads = 8 waves
    const int lane = tid & 31;
    const int wave = tid >> 5;

    __shared__ float sh_pq[256];
    __shared__ float sh_v[256];
    __shared__ float sh_e[512];
    __shared__ float red[8];

    sh_pq[tid] = pq[(size_t)bt * H + tid];
    sh_v[tid]  = v[tid];
    __syncthreads();

    const int slen = src_len[b];
    const float* peb = pe + (size_t)b * S * H;

    for (int s = wave * 64; s < wave * 64 + 64; ++s) {
        const float* per = peb + (size_t)s * H;
        float acc = 0.f;
#pragma unroll
        for (int hh = 0; hh < 256; hh += 32) {
            const int h = hh + lane;
            acc += sh_v[h] * tanhf(sh_pq[h] + per[h]);
        }
#pragma unroll
        for (int off = 16; off > 0; off >>= 1)
            acc += __shfl_xor(acc, off, 32);
        if (lane == 0)
            sh_e[s] = (s < slen) ? acc : -INFINITY;
    }
    __syncthreads();

    // ---- softmax over the 512 values in sh_e ----
    float m = -INFINITY;
    for (int s = tid; s < S; s += 256) m = fmaxf(m, sh_e[s]);
#pragma unroll
    for (int off = 16; off > 0; off >>= 1) m = fmaxf(m, __shfl_xor(m, off, 32));
    if (lane == 0) red[wave] = m;
    __syncthreads();
    m = red[0];
#pragma unroll
    for (int i = 1; i < 8; ++i) m = fmaxf(m, red[i]);
    __syncthreads();

    float sum = 0.f;
    for (int s = tid; s < S; s += 256) {
        float ex = __expf(sh_e[s] - m);   // exp(-inf - m) = 0 for masked slots
        sh_e[s] = ex;
        sum += ex;
    }
#pragma unroll
    for (int off = 16; off > 0; off >>= 1) sum += __shfl_xor(sum, off, 32);
    if (lane == 0) red[wave] = sum;
    __syncthreads();
    float total = 0.f;
#pragma unroll
    for (int i = 0; i < 8; ++i) total += red[i];
    const float inv = 1.f / total;

    for (int s = tid; s < S; s += 256)
        alpha[(size_t)bt * S + s] = sh_e[s] * inv;
}

// ---------------------------------------------------------------------------
extern "C" void kernel_launch(void* const* d_in, const int* in_sizes, int n_in,
                              void* d_out, int out_size, void* d_ws, size_t ws_size,
                              hipStream_t stream) {
    const int B = 4, T = 128, S = 512, H = 256;

    const float* query = (const float*)d_in[0];   // (B,T,H)
    const float* enc   = (const float*)d_in[1];   // (B,S,H)
    const int*   slen  = (const int*)d_in[2];     // (B)
    const float* Ws    = (const float*)d_in[3];   // (H,H)
    const float* Wh    = (const float*)d_in[4];   // (H,H)
    const float* v     = (const float*)d_in[5];   // (H)
    const float* Wout  = (const float*)d_in[6];   // (H,2H)
    float* out = (float*)d_out;                   // (B,T,H)

    float* ws    = (float*)d_ws;
    float* pq    = ws;                                  // B*T*H   = 131072
    float* pe    = pq + (size_t)B * T * H;              // B*S*H   = 524288
    float* alpha = pe + (size_t)B * S * H;              // B*T*S   = 262144
    float* ctx   = alpha + (size_t)B * T * S;           // B*T*H   = 131072
    (void)ws_size; (void)in_sizes; (void)n_in; (void)out_size;

    // 1) pq = query @ Ws^T : M=512, N=256, K=256 -> 512 tiles / 4 waves = 128 blocks
    gemm_xwT_wmma<<<128, 128, 0, stream>>>(query, Ws, pq, B * T, H, H);

    // 2) pe = enc @ Wh^T : M=2048, N=256, K=256 -> 2048 tiles -> 512 blocks
    gemm_xwT_wmma<<<512, 128, 0, stream>>>(enc, Wh, pe, B * S, H, H);

    // 3) scores + mask + softmax -> alpha
    score_softmax_kernel<<<B * T, 256, 0, stream>>>(pq, pe, v, slen, alpha);

    // 4) ctx[b] = alpha[b] @ enc[b] : per batch M=128, N=256, K=512 -> 128 tiles -> 32 blocks
    gemm_ab_batched_wmma<<<dim3(32, B), 128, 0, stream>>>(alpha, enc, ctx, T, H, S);

    // 5) out = tanh([query|ctx] @ Wout^T) : M=512, N=256, K=512 -> 512 tiles -> 128 blocks
    out_gemm_wmma<<<128, 128, 0, stream>>>(query, ctx, Wout, out);
}